// EncoderBlock_22256520528400
// MI455X (gfx1250) — compile-verified
//
#include <hip/hip_runtime.h>
#include <hip/hip_bf16.h>
#include <math.h>

typedef __attribute__((ext_vector_type(2))) float v2f;
typedef __attribute__((ext_vector_type(8))) float v8f;

#define BATCH  4
#define CIN    64
#define COUT   128
#define Lseq   4096         // H*W = 64*64
#define Di     128
#define Mtot   (BATCH*Lseq) // 16384
#define DSTATE 16
#define DTRANK 4
#define XDBL_N 36           // DT_RANK + 2*D_STATE

// ---- fp32 WMMA 16x16x4 wrapper -------------------------------------------
__device__ __forceinline__ v8f wmma4(v2f a, v2f b, v8f c) {
  // (neg_a, A, neg_b, B, c_mod, C, reuse_a, reuse_b)
  return __builtin_amdgcn_wmma_f32_16x16x4_f32(false, a, false, b, (short)0, c,
                                               false, false);
}
__device__ __forceinline__ float sigmoidf_(float x) { return 1.f / (1.f + __expf(-x)); }
__device__ __forceinline__ float siluf_(float x)    { return x * sigmoidf_(x); }

// ---- 1) in_proj: xz[m,n] = sum_k x[b,k,l] * W[n,k];  M=16384 K=64 N=256 ---
__global__ __launch_bounds__(128) void k_inproj(const float* __restrict__ x,
    const float* __restrict__ w, float* __restrict__ u_pre, float* __restrict__ z)
{
  const int lane = threadIdx.x & 31, wv = threadIdx.x >> 5;
  const int tm = blockIdx.x, tn = blockIdx.y * 4 + wv;
  const int row = lane & 15, khf = lane >> 4;
  const int m = tm * 16 + row;
  const int b = m >> 12, l = m & (Lseq - 1);
  const int n = tn * 16 + row;
  v8f acc = {};
  for (int k0 = 0; k0 < CIN; k0 += 4) {
    const int k = k0 + khf * 2;
    v2f a, bv;
    a.x  = x[(b * CIN + k) * Lseq + l];
    a.y  = x[(b * CIN + k + 1) * Lseq + l];
    bv.x = w[n * CIN + k];
    bv.y = w[n * CIN + k + 1];
    acc = wmma4(a, bv, acc);
  }
  const int col = tn * 16 + (lane & 15);
#pragma unroll
  for (int r = 0; r < 8; ++r) {
    const int mr = tm * 16 + r + 8 * khf;
    const float v = acc[r];
    if (col < Di) u_pre[mr * Di + col] = v;
    else          z[mr * Di + (col - Di)] = v;
  }
}

// ---- 2) causal depthwise conv1d (k=3, left pad 2) + bias + SiLU -----------
__global__ __launch_bounds__(256) void k_conv1d(const float* __restrict__ up,
    const float* __restrict__ cw, const float* __restrict__ cb, float* __restrict__ u)
{
  const int idx = blockIdx.x * 256 + threadIdx.x;   // m*Di + d
  const int d = idx & (Di - 1);
  const int m = idx >> 7;
  const int l = m & (Lseq - 1);
  float v = cw[d * 3 + 2] * up[idx] + cb[d];
  if (l >= 1) v += cw[d * 3 + 1] * up[idx - Di];
  if (l >= 2) v += cw[d * 3 + 0] * up[idx - 2 * Di];
  u[idx] = siluf_(v);
}

// ---- 3) x_proj: xdbl[m,n<36] = sum_k u[m,k]*W[n,k]; N padded to 48 --------
__global__ __launch_bounds__(96) void k_xproj(const float* __restrict__ u,
    const float* __restrict__ w, float* __restrict__ xdbl)
{
  const int lane = threadIdx.x & 31;
  const int tn = threadIdx.x >> 5;                  // 0..2
  const int tm = blockIdx.x;
  const int row = lane & 15, khf = lane >> 4;
  const int m = tm * 16 + row;
  const int n = tn * 16 + row;
  v8f acc = {};
  for (int k0 = 0; k0 < Di; k0 += 4) {
    const int k = k0 + khf * 2;
    v2f a, bv;
    a.x  = u[m * Di + k];
    a.y  = u[m * Di + k + 1];
    bv.x = (n < XDBL_N) ? w[n * Di + k]     : 0.f;
    bv.y = (n < XDBL_N) ? w[n * Di + k + 1] : 0.f;
    acc = wmma4(a, bv, acc);
  }
  const int col = tn * 16 + (lane & 15);
  if (col < XDBL_N) {
#pragma unroll
    for (int r = 0; r < 8; ++r) {
      const int mr = tm * 16 + r + 8 * khf;
      xdbl[mr * XDBL_N + col] = acc[r];
    }
  }
}

// ---- 4) dt = softplus(dt_raw @ dt_proj_w.T + dt_proj_b) -------------------
__global__ __launch_bounds__(256) void k_dtproj(const float* __restrict__ xdbl,
    const float* __restrict__ dtw, const float* __restrict__ dtb, float* __restrict__ dt)
{
  const int idx = blockIdx.x * 256 + threadIdx.x;   // m*Di + d
  const int d = idx & (Di - 1);
  const int m = idx >> 7;
  float v = dtb[d];
#pragma unroll
  for (int r = 0; r < DTRANK; ++r)
    v += xdbl[m * XDBL_N + r] * dtw[d * DTRANK + r];
  dt[idx] = (v > 20.f) ? v : log1pf(__expf(v));
}

// ---- 5) selective scan: 512 channels x 16 state-lanes, sequential in L ----
__global__ __launch_bounds__(256) void k_scan(const float* __restrict__ dt,
    const float* __restrict__ xdbl, const float* __restrict__ u,
    const float* __restrict__ z, const float* __restrict__ A_log,
    const float* __restrict__ Dp, float* __restrict__ y)
{
  const int lane16 = threadIdx.x & 15;
  const int grp = blockIdx.x * 16 + (threadIdx.x >> 4); // 0..511 = (b,d)
  const int b = grp >> 7, d = grp & 127;
  const float a  = -__expf(A_log[d * DSTATE + lane16]);
  const float dp = Dp[d];
  const int base = b * Lseq;
  float h = 0.f;
  for (int l = 0; l < Lseq; ++l) {
    const int mi = base + l;
    if (l + 8 < Lseq) {                                  // global_prefetch_b8
      __builtin_prefetch(&dt[(mi + 8) * Di + d], 0, 1);
      __builtin_prefetch(&xdbl[(mi + 8) * XDBL_N + lane16], 0, 1);
    }
    const float dtv = dt[mi * Di + d];
    const float uv  = u[mi * Di + d];
    const float Bv  = xdbl[mi * XDBL_N + DTRANK + lane16];
    const float Cv  = xdbl[mi * XDBL_N + DTRANK + DSTATE + lane16];
    const float dA  = __expf(dtv * a);
    h = dA * h + dtv * Bv * uv;
    float p = h * Cv;
    p += __shfl_xor(p, 8, 16);
    p += __shfl_xor(p, 4, 16);
    p += __shfl_xor(p, 2, 16);
    p += __shfl_xor(p, 1, 16);
    if (lane16 == 0) {
      const float zv = z[mi * Di + d];
      y[mi * Di + d] = (p + dp * uv) * (zv * sigmoidf_(zv));
    }
  }
}

// ---- 6) out_proj GEMM + residual: x1 = x + y @ W.T;  K=128 N=64 -----------
__global__ __launch_bounds__(128) void k_outproj(const float* __restrict__ y,
    const float* __restrict__ w, const float* __restrict__ x, float* __restrict__ x1)
{
  const int lane = threadIdx.x & 31;
  const int tn = threadIdx.x >> 5;                  // 0..3
  const int tm = blockIdx.x;
  const int row = lane & 15, khf = lane >> 4;
  const int m = tm * 16 + row;
  const int n = tn * 16 + row;
  v8f acc = {};
  for (int k0 = 0; k0 < Di; k0 += 4) {
    const int k = k0 + khf * 2;
    v2f a, bv;
    a.x  = y[m * Di + k];     a.y  = y[m * Di + k + 1];
    bv.x = w[n * Di + k];     bv.y = w[n * Di + k + 1];
    acc = wmma4(a, bv, acc);
  }
  const int col = tn * 16 + (lane & 15);
  const int b = (tm * 16) >> 12;                    // tile never crosses batch
#pragma unroll
  for (int r = 0; r < 8; ++r) {
    const int mr = tm * 16 + r + 8 * khf;
    const int l  = mr & (Lseq - 1);
    const int gi = (b * CIN + col) * Lseq + l;
    x1[gi] = x[gi] + acc[r];
  }
}

// ---- 7) conv3x3 SAME + BN + ReLU: implicit GEMM, K=576, N=128 -------------
// One block = one 16-position strip (single image row). 8 waves cover all 8
// N-tiles and share a 3x18x64 halo patch staged in LDS via async-to-LDS DMA.
// Reduction order k' = (kh*3+kw)*64 + ci so A-fragments are aligned b64 LDS
// reads (channel-fastest) and need no per-step bounds checks.
__global__ __launch_bounds__(256) void k_conv3x3(const float* __restrict__ x1,
    const float* __restrict__ pw, const float* __restrict__ pb,
    const float* __restrict__ gamma, const float* __restrict__ beta,
    const float* __restrict__ mean, const float* __restrict__ var,
    float* __restrict__ skip)
{
  __shared__ float patch[3 * 18 * 64];              // 13824 B
  const int tid = threadIdx.x;
  const int lane = tid & 31, wv = tid >> 5;         // wv = N-tile 0..7
  const int tm = blockIdx.x;
  const int row = lane & 15, khf = lane >> 4;
  const int m0 = tm * 16;
  const int b = m0 >> 12, l0 = m0 & (Lseq - 1);
  const int h = l0 >> 6, w0 = l0 & 63;
  const int n = wv * 16 + row;

  // SAME-padding zeros first (out-of-bounds elements keep them)
  for (int e = tid; e < 3456; e += 256) patch[e] = 0.f;
  __syncthreads();
  // async DMA the in-bounds halo elements straight into LDS
  for (int e = tid; e < 3456; e += 256) {
    const int ci = e & 63;
    const int t  = e >> 6;                          // row*18 + ww
    const int rr = t / 18;
    const int ww = t - rr * 18;
    const int hi = h + rr - 1, wi = w0 + ww - 1;
    if ((unsigned)hi < 64u && (unsigned)wi < 64u) {
      const float* gp = &x1[((b * CIN + ci) << 12) + (hi << 6) + wi];
      const unsigned la = (unsigned)(size_t)&patch[e];
      asm volatile("global_load_async_to_lds_b32 %0, %1, off"
                   :: "v"(la), "v"((unsigned long long)(size_t)gp)
                   : "memory");
    }
  }
  asm volatile("s_wait_asynccnt 0x0" ::: "memory");
  __syncthreads();

  v8f acc = {};
#pragma unroll
  for (int kh = 0; kh < 3; ++kh) {
#pragma unroll
    for (int kw = 0; kw < 3; ++kw) {
      const int abase = (kh * 18 + row + kw) * 64;  // per-lane A row base
      const int wb = n * 576 + kh * 3 + kw;         // B base for this tap
      for (int c0 = 0; c0 < 64; c0 += 4) {
        const int ci = c0 + khf * 2;
        const v2f a = *(const v2f*)&patch[abase + ci];  // ds_load_b64
        v2f bv;
        bv.x = pw[wb + ci * 9];
        bv.y = pw[wb + ci * 9 + 9];
        acc = wmma4(a, bv, acc);
      }
    }
  }

  const int col = wv * 16 + (lane & 15);
  const float rs  = rsqrtf(var[col] + 1e-5f);
  const float sc  = rs * gamma[col];
  const float off = (pb[col] - mean[col]) * sc + beta[col];
#pragma unroll
  for (int r = 0; r < 8; ++r) {
    const int mr = m0 + r + 8 * khf;
    const int lr = mr & (Lseq - 1);
    float v = acc[r] * sc + off;
    skip[((b * COUT + col) << 12) + lr] = v > 0.f ? v : 0.f;
  }
}

// ---- 8) 2x2 max + avg pool ------------------------------------------------
__global__ __launch_bounds__(256) void k_pool(const float* __restrict__ skip,
    float* __restrict__ mp, float* __restrict__ ap)
{
  const int idx = blockIdx.x * 256 + threadIdx.x;   // (b*COUT+c)*1024 + p
  const int p = idx & 1023;
  const int bc = idx >> 10;
  const int ph = p >> 5, pw_ = p & 31;
  const int s = (bc << 12) + ((ph * 2) << 6) + pw_ * 2;
  const float v00 = skip[s],      v01 = skip[s + 1];
  const float v10 = skip[s + 64], v11 = skip[s + 65];
  mp[idx] = fmaxf(fmaxf(v00, v01), fmaxf(v10, v11));
  ap[idx] = 0.25f * (v00 + v01 + v10 + v11);
}

// ---- 9) gate 1x1 GEMM (K=256 over concat[mp,ap]) + sigmoid blend ----------
__global__ __launch_bounds__(128) void k_gate(const float* __restrict__ mp,
    const float* __restrict__ ap, const float* __restrict__ gw,
    const float* __restrict__ gb, float* __restrict__ down)
{
  const int lane = threadIdx.x & 31, wv = threadIdx.x >> 5;
  const int tm = blockIdx.x;                         // M = 4096 -> 256 tiles
  const int tn = blockIdx.y * 4 + wv;                // 0..7
  const int row = lane & 15, khf = lane >> 4;
  const int m = tm * 16 + row;
  const int b = m >> 10, p = m & 1023;
  const int n = tn * 16 + row;
  v8f acc = {};
  for (int k0 = 0; k0 < 2 * COUT; k0 += 4) {
    const int k = k0 + khf * 2;                      // even: pair stays in mp or ap
    v2f a, bv;
    a.x  = (k < COUT) ? mp[((b * COUT + k) << 10) + p]
                      : ap[((b * COUT + k - COUT) << 10) + p];
    a.y  = (k + 1 < COUT) ? mp[((b * COUT + k + 1) << 10) + p]
                          : ap[((b * COUT + k + 1 - COUT) << 10) + p];
    bv.x = gw[n * (2 * COUT) + k];
    bv.y = gw[n * (2 * COUT) + k + 1];
    acc = wmma4(a, bv, acc);
  }
  const int col = tn * 16 + (lane & 15);
  const float gbv = gb[col];
  const int bb = (tm * 16) >> 10;
#pragma unroll
  for (int r = 0; r < 8; ++r) {
    const int mr = tm * 16 + r + 8 * khf;
    const int pr = mr & 1023;
    const int gi = ((bb * COUT + col) << 10) + pr;
    const float g = sigmoidf_(acc[r] + gbv);
    down[gi] = g * mp[gi] + (1.f - g) * ap[gi];
  }
}

// ---------------------------------------------------------------------------
extern "C" void kernel_launch(void* const* d_in, const int* in_sizes, int n_in,
                              void* d_out, int out_size, void* d_ws, size_t ws_size,
                              hipStream_t stream) {
  (void)in_sizes; (void)n_in; (void)out_size; (void)ws_size;
  const float* x    = (const float*)d_in[0];
  const float* inW  = (const float*)d_in[1];
  const float* c1w  = (const float*)d_in[2];
  const float* c1b  = (const float*)d_in[3];
  const float* xpW  = (const float*)d_in[4];
  const float* dtW  = (const float*)d_in[5];
  const float* dtB  = (const float*)d_in[6];
  const float* Alog = (const float*)d_in[7];
  const float* Dp   = (const float*)d_in[8];
  const float* outW = (const float*)d_in[9];
  const float* pwW  = (const float*)d_in[10];
  const float* pwB  = (const float*)d_in[11];
  const float* bnG  = (const float*)d_in[12];
  const float* bnB  = (const float*)d_in[13];
  const float* bnM  = (const float*)d_in[14];
  const float* bnV  = (const float*)d_in[15];
  const float* gW   = (const float*)d_in[16];
  const float* gB   = (const float*)d_in[17];

  float* out = (float*)d_out;
  float* down = out;                           // (4,128,32,32) = 524288
  float* skip = out + 524288;                  // (4,128,64,64) = 2097152

  // workspace layout (floats); total ~11.08M floats = 44.3 MB
  const size_t N_MD = (size_t)Mtot * Di;       // 2097152
  float* ws    = (float*)d_ws;
  float* u_pre = ws;
  float* zbuf  = ws + N_MD;
  float* ubuf  = ws + 2 * N_MD;
  float* dtbuf = ws + 3 * N_MD;
  float* xdbl  = ws + 4 * N_MD;                        // Mtot*36
  float* x1    = xdbl + (size_t)Mtot * XDBL_N;         // 1048576
  float* mp    = x1 + (size_t)BATCH * CIN * Lseq;      // 524288
  float* ap    = mp + 524288;                          // 524288
  float* ybuf  = u_pre;                                // alias: u_pre dead by then

  k_inproj <<<dim3(Mtot / 16, 4), 128, 0, stream>>>(x, inW, u_pre, zbuf);
  k_conv1d <<<N_MD / 256, 256, 0, stream>>>(u_pre, c1w, c1b, ubuf);
  k_xproj  <<<Mtot / 16, 96, 0, stream>>>(ubuf, xpW, xdbl);
  k_dtproj <<<N_MD / 256, 256, 0, stream>>>(xdbl, dtW, dtB, dtbuf);
  k_scan   <<<32, 256, 0, stream>>>(dtbuf, xdbl, ubuf, zbuf, Alog, Dp, ybuf);
  k_outproj<<<Mtot / 16, 128, 0, stream>>>(ybuf, outW, x, x1);
  k_conv3x3<<<Mtot / 16, 256, 0, stream>>>(x1, pwW, pwB, bnG, bnB, bnM, bnV, skip);
  k_pool   <<<524288 / 256, 256, 0, stream>>>(skip, mp, ap);
  k_gate   <<<dim3(4096 / 16, 2), 128, 0, stream>>>(mp, ap, gW, gB, down);
}